// DenseFlashAttention_16123307229343
// MI455X (gfx1250) — compile-verified
//
#include <hip/hip_runtime.h>
#include <hip/hip_bf16.h>

// ---------------------------------------------------------------------------
// Graph attention (radial+tangential segment softmax) for MI455X / gfx1250.
//  - GEMMs: native f32 WMMA (v_wmma_f32_16x16x4_f32), 16-node x 64-col tile
//    per wave, weights staged in LDS via gfx1250 async global->LDS copies.
//  - Edge phase: wave-per-edge coalesced gathers + coalesced f32 atomics
//    (entire working set is L2-resident: 192 MB L2 >> 120 MB scratch).
// ---------------------------------------------------------------------------

#define FEAT   64
#define HEADS  4

typedef __attribute__((ext_vector_type(2))) float v2f;
typedef __attribute__((ext_vector_type(8))) float v8f;

// pointer types matching __builtin_amdgcn_global_load_async_to_lds_b128:
// (int4 __device__*, int4 __shared__*, imm offset, imm cpol)
typedef int v4i_vs __attribute__((vector_size(16)));
typedef __attribute__((address_space(1))) v4i_vs* async_gptr;
typedef __attribute__((address_space(3))) v4i_vs* async_lptr;

// ---- order-preserving float <-> uint for atomic max over signed floats ----
__device__ __forceinline__ unsigned ford(float f) {
    unsigned u = __float_as_uint(f);
    return (u & 0x80000000u) ? ~u : (u | 0x80000000u);
}
__device__ __forceinline__ float deord(unsigned o) {
    return __uint_as_float((o & 0x80000000u) ? (o & 0x7FFFFFFFu) : ~o);
}
__device__ __forceinline__ float nan2num(float v) {
    if (isnan(v)) return 0.0f;
    if (isinf(v)) return v > 0.0f ? 3.4028234663852886e38f : -3.4028234663852886e38f;
    return v;
}

// ---- stage a 64x64 f32 weight matrix into LDS -----------------------------
// Uses gfx1250 async global->LDS copies (tracked by ASYNCcnt, waited with
// s_wait_asynccnt) when the builtin exists, else a plain float4 copy.
__device__ __forceinline__ void stage_weights(const float* __restrict__ src,
                                              float* dst_lds,
                                              int tid, int nthreads) {
#if __has_builtin(__builtin_amdgcn_global_load_async_to_lds_b128)
    for (int i = tid * 4; i < FEAT * FEAT; i += nthreads * 4) {
        __builtin_amdgcn_global_load_async_to_lds_b128(
            (async_gptr)(src + i),
            (async_lptr)(dst_lds + i),
            0, 0);
    }
#if __has_builtin(__builtin_amdgcn_s_wait_asynccnt)
    __builtin_amdgcn_s_wait_asynccnt(0);
#else
    asm volatile("s_wait_asynccnt 0x0" ::: "memory");
#endif
#else
    const float4* s4 = (const float4*)src;
    float4* d4 = (float4*)dst_lds;
    for (int i = tid; i < FEAT * FEAT / 4; i += nthreads)
        d4[i] = s4[i];
#endif
}

// ---------------------------------------------------------------------------
// Kernel 1: per-head projection  proj[n,h,g] = sum_f x[n,f] * w_proj[h,f,g]
// One wave = 16-node x 64-out tile for one head.
// ---------------------------------------------------------------------------
__global__ void proj_wmma_kernel(const float* __restrict__ x,
                                 const float* __restrict__ w_proj,
                                 float* __restrict__ proj,
                                 int n_nodes)
{
    __shared__ float w_lds[FEAT * FEAT];
    const int head = blockIdx.y;
    stage_weights(w_proj + (size_t)head * FEAT * FEAT, w_lds, threadIdx.x, blockDim.x);
    __syncthreads();

    const int wave = threadIdx.x >> 5;
    const int lane = threadIdx.x & 31;
    const int node_base = (blockIdx.x * (blockDim.x >> 5) + wave) * 16;
    if (node_base >= n_nodes) return;           // wave-uniform: EXEC stays all-1s

    const int row = lane & 15;                  // M index (A) / N index (B,D)
    const int khi = lane >> 4;                  // 0: K=0,1   1: K=2,3
    int arow = node_base + row;
    if (arow >= n_nodes) arow = n_nodes - 1;    // clamp (stores are guarded)

    v8f acc[4];
    const v8f vzero = {0.f,0.f,0.f,0.f,0.f,0.f,0.f,0.f};
#pragma unroll
    for (int nt = 0; nt < 4; ++nt) acc[nt] = vzero;

    const float* xr = x + (size_t)arow * FEAT;
#pragma unroll
    for (int k = 0; k < FEAT; k += 4) {
        v2f a;
        a.x = xr[k + 2 * khi];
        a.y = xr[k + 2 * khi + 1];
#pragma unroll
        for (int nt = 0; nt < 4; ++nt) {
            const int col = nt * 16 + row;
            v2f b;
            b.x = w_lds[(k + 2 * khi) * FEAT + col];
            b.y = w_lds[(k + 2 * khi + 1) * FEAT + col];
            acc[nt] = __builtin_amdgcn_wmma_f32_16x16x4_f32(
                false, a, false, b, (short)0, acc[nt], false, false);
        }
    }

    // D layout: VGPR r -> M = r + 8*khi, N = lane&15
#pragma unroll
    for (int nt = 0; nt < 4; ++nt) {
#pragma unroll
        for (int r = 0; r < 8; ++r) {
            const int node = node_base + r + 8 * khi;
            if (node < n_nodes)
                proj[(size_t)node * HEADS * FEAT + (size_t)head * FEAT + nt * 16 + row] = acc[nt][r];
        }
    }
}

// ---------------------------------------------------------------------------
// Kernel 2: wave-per-edge logits + segment atomic-max.
// Lane l owns 8 contiguous feats of the 256-float (H*F) row; head = l>>3.
// Coalesced 1KB gathers of proj[s], proj[r]; shfl reduction per head group.
// ---------------------------------------------------------------------------
__global__ void edge_logits_kernel(const int* __restrict__ ei,
                                   const float* __restrict__ edge_len,
                                   const float* __restrict__ proj,
                                   const float* __restrict__ rscore,
                                   const float* __restrict__ tscore,
                                   const float* __restrict__ rdscale,
                                   float* __restrict__ logit_r,
                                   float* __restrict__ logit_t,
                                   unsigned* __restrict__ m_r,
                                   unsigned* __restrict__ m_t,
                                   int n_edges)
{
    const int e    = (blockIdx.x * blockDim.x + threadIdx.x) >> 5;
    const int lane = threadIdx.x & 31;
    if (e >= n_edges) return;                   // wave-uniform

    const int s = ei[e];
    const int r = ei[n_edges + e];
    const int h      = lane >> 3;               // head handled by this lane
    const int fbase  = lane * 8;                // offset within the 256-float row
    const int flocal = fbase & (FEAT - 1);      // offset within the head

    const float* ps = proj + (size_t)s * HEADS * FEAT + fbase;
    const float* pr = proj + (size_t)r * HEADS * FEAT + fbase;
    const float* rs = rscore + h * FEAT + flocal;
    const float* ts = tscore + h * FEAT + flocal;

    float dr = 0.f, dt = 0.f;
#pragma unroll
    for (int j = 0; j < 8; j += 4) {
        const float4 a  = *(const float4*)(ps + j);
        const float4 b  = *(const float4*)(pr + j);
        const float4 rw = *(const float4*)(rs + j);
        const float4 tw = *(const float4*)(ts + j);
        const float dx = a.x - b.x, dy = a.y - b.y, dz = a.z - b.z, dw = a.w - b.w;
        dr += dx * rw.x + dy * rw.y + dz * rw.z + dw * rw.w;
        dt += dx * tw.x + dy * tw.y + dz * tw.z + dw * tw.w;
    }
    // reduce within each aligned 8-lane head group
#pragma unroll
    for (int m = 4; m >= 1; m >>= 1) {
        dr += __shfl_xor(dr, m, 32);
        dt += __shfl_xor(dt, m, 32);
    }

    if ((lane & 7) == 0) {
        const float rl = dr - rdscale[0] * edge_len[e];
        const float tl = dt;
        logit_r[e * HEADS + h] = rl;
        logit_t[e * HEADS + h] = tl;
        atomicMax(&m_r[r * HEADS + h], ford(rl));
        atomicMax(&m_t[r * HEADS + h], ford(tl));
    }
}

// ---------------------------------------------------------------------------
// Kernel 3: e = exp(l - m[recv]) (in place), atomic segment sum.
// ---------------------------------------------------------------------------
__global__ void exp_sum_kernel(const int* __restrict__ ei,
                               float* __restrict__ logit_r,
                               float* __restrict__ logit_t,
                               const unsigned* __restrict__ m_r,
                               const unsigned* __restrict__ m_t,
                               float* __restrict__ s_r,
                               float* __restrict__ s_t,
                               int n_edges)
{
    const int idx = blockIdx.x * blockDim.x + threadIdx.x;
    if (idx >= n_edges * HEADS) return;
    const int e = idx >> 2;
    const int h = idx & 3;
    const int r = ei[n_edges + e];

    const float er = __expf(logit_r[idx] - deord(m_r[r * HEADS + h]));
    const float et = __expf(logit_t[idx] - deord(m_t[r * HEADS + h]));
    logit_r[idx] = er;
    logit_t[idx] = et;
    atomicAdd(&s_r[r * HEADS + h], er);
    atomicAdd(&s_t[r * HEADS + h], et);
}

// ---------------------------------------------------------------------------
// Kernel 4: wave-per-edge scatter of (alpha_r + alpha_t) * delta into
// acc[recv]. Coalesced gathers + coalesced global_atomic_add_f32 bursts.
// ---------------------------------------------------------------------------
__global__ void scatter_kernel(const int* __restrict__ ei,
                               const float* __restrict__ proj,
                               const float* __restrict__ e_r,
                               const float* __restrict__ e_t,
                               const float* __restrict__ s_r,
                               const float* __restrict__ s_t,
                               float* __restrict__ acc,
                               int n_edges)
{
    const int e    = (blockIdx.x * blockDim.x + threadIdx.x) >> 5;
    const int lane = threadIdx.x & 31;
    if (e >= n_edges) return;                   // wave-uniform

    const int s = ei[e];
    const int r = ei[n_edges + e];
    const int h     = lane >> 3;
    const int fbase = lane * 8;

    const float w = e_r[e * HEADS + h] / s_r[r * HEADS + h]
                  + e_t[e * HEADS + h] / s_t[r * HEADS + h];

    const float* ps = proj + (size_t)s * HEADS * FEAT + fbase;
    const float* pr = proj + (size_t)r * HEADS * FEAT + fbase;
    float*       ao = acc  + (size_t)r * HEADS * FEAT + fbase;
#pragma unroll
    for (int j = 0; j < 8; ++j)
        atomicAdd(&ao[j], w * (ps[j] - pr[j]));
}

// ---------------------------------------------------------------------------
// Kernel 5: y = x + nan2num(mean_h acc) @ w_out     (f32 WMMA again)
// ---------------------------------------------------------------------------
__global__ void out_wmma_kernel(const float* __restrict__ x,
                                const float* __restrict__ acc_in,
                                const float* __restrict__ w_out,
                                float* __restrict__ y,
                                int n_nodes)
{
    __shared__ float w_lds[FEAT * FEAT];
    stage_weights(w_out, w_lds, threadIdx.x, blockDim.x);
    __syncthreads();

    const int wave = threadIdx.x >> 5;
    const int lane = threadIdx.x & 31;
    const int node_base = (blockIdx.x * (blockDim.x >> 5) + wave) * 16;
    if (node_base >= n_nodes) return;           // wave-uniform

    const int row = lane & 15;
    const int khi = lane >> 4;
    int arow = node_base + row;
    if (arow >= n_nodes) arow = n_nodes - 1;

    v8f acc[4];
    const v8f vzero = {0.f,0.f,0.f,0.f,0.f,0.f,0.f,0.f};
#pragma unroll
    for (int nt = 0; nt < 4; ++nt) acc[nt] = vzero;

    const float* ar = acc_in + (size_t)arow * HEADS * FEAT;
#pragma unroll
    for (int k = 0; k < FEAT; k += 4) {
        const int f0 = k + 2 * khi;
        v2f a;
        a.x = nan2num(0.25f * (ar[f0]     + ar[FEAT + f0]     + ar[2*FEAT + f0]     + ar[3*FEAT + f0]));
        a.y = nan2num(0.25f * (ar[f0 + 1] + ar[FEAT + f0 + 1] + ar[2*FEAT + f0 + 1] + ar[3*FEAT + f0 + 1]));
#pragma unroll
        for (int nt = 0; nt < 4; ++nt) {
            const int col = nt * 16 + row;
            v2f b;
            b.x = w_lds[f0 * FEAT + col];
            b.y = w_lds[(f0 + 1) * FEAT + col];
            acc[nt] = __builtin_amdgcn_wmma_f32_16x16x4_f32(
                false, a, false, b, (short)0, acc[nt], false, false);
        }
    }

#pragma unroll
    for (int nt = 0; nt < 4; ++nt) {
#pragma unroll
        for (int r = 0; r < 8; ++r) {
            const int node = node_base + r + 8 * khi;
            if (node < n_nodes) {
                const size_t o = (size_t)node * FEAT + nt * 16 + row;
                y[o] = x[o] + acc[nt][r];
            }
        }
    }
}

// ---------------------------------------------------------------------------
static inline size_t align256(size_t v) { return (v + 255) & ~(size_t)255; }

extern "C" void kernel_launch(void* const* d_in, const int* in_sizes, int n_in,
                              void* d_out, int out_size, void* d_ws, size_t ws_size,
                              hipStream_t stream) {
    const float* x        = (const float*)d_in[0];
    const int*   ei       = (const int*)  d_in[1];   // [2, E]
    /* d_in[2] = edge_vec: unused by the reference forward */
    const float* edge_len = (const float*)d_in[3];
    const float* w_proj   = (const float*)d_in[4];   // [H, F, F]
    const float* rscore   = (const float*)d_in[5];   // [H, F]
    const float* tscore   = (const float*)d_in[6];   // [H, F]
    const float* rdscale  = (const float*)d_in[7];   // scalar
    const float* w_out    = (const float*)d_in[8];   // [F, F]
    float*       y        = (float*)d_out;

    const int n_nodes = in_sizes[0] / FEAT;
    const int n_edges = in_sizes[1] / 2;

    // -------- workspace carve (all offsets 256B aligned) --------
    const size_t projB = (size_t)n_nodes * HEADS * FEAT * sizeof(float);
    const size_t logB  = (size_t)n_edges * HEADS * sizeof(float);
    const size_t cntB  = (size_t)n_nodes * HEADS * sizeof(float);

    char* p = (char*)d_ws;
    float*    proj    = (float*)p;    p += align256(projB);
    float*    logit_r = (float*)p;    p += align256(logB);
    float*    logit_t = (float*)p;    p += align256(logB);
    // zero-initialized region starts here:
    char*     zstart  = p;
    float*    acc     = (float*)p;    p += align256(projB);
    unsigned* m_r     = (unsigned*)p; p += align256(cntB);
    unsigned* m_t     = (unsigned*)p; p += align256(cntB);
    float*    s_r     = (float*)p;    p += align256(cntB);
    float*    s_t     = (float*)p;    p += align256(cntB);
    char*     zend    = p;

    // zero acc / maxes (ord(x) >= 0 for all reachable logits) / sums, per call
    (void)hipMemsetAsync(zstart, 0, (size_t)(zend - zstart), stream);

    // -------- 1: per-head projection (WMMA) --------
    const int tiles16 = (n_nodes + 15) / 16;
    {
        dim3 grid((tiles16 + 3) / 4, HEADS);
        proj_wmma_kernel<<<grid, 128, 0, stream>>>(x, w_proj, proj, n_nodes);
    }

    // -------- 2..4: edge phase --------
    const int wave_blocks = (n_edges + 7) / 8;              // 8 waves (256 thr) / block
    edge_logits_kernel<<<wave_blocks, 256, 0, stream>>>(ei, edge_len, proj, rscore, tscore,
                                                        rdscale, logit_r, logit_t, m_r, m_t, n_edges);
    const int eh = n_edges * HEADS;
    exp_sum_kernel<<<(eh + 255) / 256, 256, 0, stream>>>(ei, logit_r, logit_t, m_r, m_t,
                                                         s_r, s_t, n_edges);
    scatter_kernel<<<wave_blocks, 256, 0, stream>>>(ei, proj, logit_r, logit_t,
                                                    s_r, s_t, acc, n_edges);

    // -------- 5: mean/nan2num + output GEMM + residual (WMMA) --------
    out_wmma_kernel<<<(tiles16 + 3) / 4, 128, 0, stream>>>(x, acc, w_out, y, n_nodes);
}